// Radon_6425271074978
// MI455X (gfx1250) — compile-verified
//
#include <hip/hip_runtime.h>
#include <stdint.h>

// Problem constants (from reference): B=8, N=256, A=180
#define NN      256
#define A_ANG   180
#define B_BATCH 8

typedef float v2f __attribute__((ext_vector_type(2)));
typedef float v8f __attribute__((ext_vector_type(8)));
typedef unsigned int v4u __attribute__((ext_vector_type(4)));
typedef int v4i __attribute__((ext_vector_type(4)));
typedef int v8i __attribute__((ext_vector_type(8)));

__global__ __launch_bounds__(256)
void radon_fwd_kernel(const float* __restrict__ imgs,
                      const float* __restrict__ angles,
                      float* __restrict__ out)
{
    // Full fp32 image in LDS: 256*256*4 = 256 KB (WGP budget is 320 KB)
    __shared__ float img_s[NN * NN];

    const int a   = blockIdx.x;   // angle
    const int b   = blockIdx.y;   // batch
    const int tid = threadIdx.x;

    // ---------------- TDM: async tensor load of image b into LDS -------------
    if (tid < 32) { // wave 0 issues exactly one TDM op
        const uint64_t gaddr = (uint64_t)(uintptr_t)(const void*)(imgs + (size_t)b * NN * NN);
        const uint32_t lds   = (uint32_t)(uintptr_t)(void*)img_s; // LDS byte offset

        v4u g0;
        g0[0] = 1u;                                   // count=1 (valid), user mode
        g0[1] = lds;                                  // lds_addr
        g0[2] = (uint32_t)(gaddr & 0xFFFFFFFFu);      // global_addr[31:0]
        g0[3] = (uint32_t)((gaddr >> 32) & 0x1FFFFFFu)// global_addr[56:32]
              | (2u << 30);                           // type = 2 ("image")

        v8i g1;
        g1[0] = (int)(2u << 16);                      // data_size=2 -> 4 bytes; mask=0
        g1[1] = (int)((uint32_t)NN << 16);            // tensor_dim0[15:0]=256
        g1[2] = (int)((uint32_t)NN << 16);            // dim0 hi=0 | tensor_dim1[15:0]=256
        g1[3] = (int)((uint32_t)NN << 16);            // dim1 hi=0 | tile_dim0=256
        g1[4] = (int)NN;                              // tile_dim1=256, tile_dim2=0
        g1[5] = (int)NN;                              // tensor_dim0_stride = 256
        g1[6] = 0;
        g1[7] = 0;

        v4i gz4 = {0, 0, 0, 0};                       // 2-D tensor: groups 2/3 unused
        v8i gz8 = {0, 0, 0, 0, 0, 0, 0, 0};
        __builtin_amdgcn_tensor_load_to_lds(g0, g1, gz4, gz4, gz8, 0);
        __builtin_amdgcn_s_wait_tensorcnt(0);
    }
    __syncthreads();

    // ---------------- ray integration --------------------------------------
    const float ang = angles[a];
    float sa, ca;
    sincosf(ang, &sa, &ca);

    const float cc   = (float)(NN - 1) * 0.5f;  // 127.5
    const int   lane = tid & 31;
    const int   wave = tid >> 5;                // 8 waves
    const int   m    = lane & 15;               // detector within 16-wide chunk
    const int   half = lane >> 4;               // K-half of the t-integration

    // each wave handles 2 chunks of 16 detectors -> 8 waves cover s = 0..255
    for (int sc = wave; sc < 16; sc += 8) {
        const int   s_idx = sc * 16 + m;
        const float sv = (float)s_idx - cc;
        const float xs = sv * ca + cc;          // x at (t - c) = 0
        const float ys = sv * sa + cc;          // y at (t - c) = 0

        // B-matrix (4x16): column = lane%16 (detector); the 4 K-slots hold the
        // 4 t-chunk partials (placement across (VGPR, lane-half) is summed out).
        v2f acc; acc[0] = 0.0f; acc[1] = 0.0f;

#pragma unroll
        for (int j = 0; j < 2; ++j) {
            const int t0 = half * 128 + j * 64; // 64-sample t chunk per accumulator
            float av = 0.0f;
            for (int tt = 0; tt < 64; ++tt) {
                const float tv = (float)(t0 + tt) - cc;
                const float x  = xs - tv * sa;
                const float y  = ys + tv * ca;
                const float x0f = floorf(x);
                const float y0f = floorf(y);
                const float wx  = x - x0f;
                const float wy  = y - y0f;
                const int x0 = (int)x0f, y0 = (int)y0f;
                const int x1 = x0 + 1,   y1 = y0 + 1;

                const int x0c = min(max(x0, 0), NN - 1);
                const int x1c = min(max(x1, 0), NN - 1);
                const int y0c = min(max(y0, 0), NN - 1);
                const int y1c = min(max(y1, 0), NN - 1);

                // 1-D weights with the bounds mask folded in (one cndmask each)
                const float ax0 = ((unsigned)x0 < (unsigned)NN) ? (1.0f - wx) : 0.0f;
                const float ax1 = ((unsigned)x1 < (unsigned)NN) ? wx          : 0.0f;
                const float by0 = ((unsigned)y0 < (unsigned)NN) ? (1.0f - wy) : 0.0f;
                const float by1 = ((unsigned)y1 < (unsigned)NN) ? wy          : 0.0f;

                const int r0 = y0c * NN;
                const int r1 = y1c * NN;
                const float v00 = img_s[r0 + x0c];
                const float v10 = img_s[r0 + x1c];
                const float v01 = img_s[r1 + x0c];
                const float v11 = img_s[r1 + x1c];

                const float row0 = fmaf(ax1, v10, ax0 * v00);
                const float row1 = fmaf(ax1, v11, ax0 * v01);
                av = fmaf(by0, row0, av);
                av = fmaf(by1, row1, av);
            }
            acc[j] = av;
        }

        // Cross-lane reduction on the matrix unit (A = ones):
        //   D[m,n] = sum_k 1 * B[k,n] = total[detector n], replicated over m.
        // => every lane holds total[lane%16] in every D register.
        v2f ones; ones[0] = 1.0f; ones[1] = 1.0f;
        v8f cz = {};
        v8f d = __builtin_amdgcn_wmma_f32_16x16x4_f32(
                    /*neg_a=*/false, ones, /*neg_b=*/false, acc,
                    /*c_mod=*/(short)0, cz, /*reuse_a=*/false, /*reuse_b=*/false);

        // Single coalesced 16-wide store (lanes 16..31 hold duplicates).
        float* orow = out + ((size_t)b * A_ANG + a) * NN + sc * 16;
        if (lane < 16) {
            orow[lane] = d[0];
        }
    }
}

extern "C" void kernel_launch(void* const* d_in, const int* in_sizes, int n_in,
                              void* d_out, int out_size, void* d_ws, size_t ws_size,
                              hipStream_t stream) {
    const float* imgs   = (const float*)d_in[0];  // [B, N, N] fp32
    const float* angles = (const float*)d_in[1];  // [A] fp32
    float* out = (float*)d_out;                   // [B, A, N] fp32

    dim3 grid(A_ANG, B_BATCH);
    radon_fwd_kernel<<<grid, 256, 0, stream>>>(imgs, angles, out);
}